// FEMRMamba_82944408420550
// MI455X (gfx1250) — compile-verified
//
#include <hip/hip_runtime.h>
#include <hip/hip_bf16.h>
#include <math.h>
#include <stdint.h>

// ---------------- static model config ----------------
#define D_MODEL 768
#define D_INNER 1536
#define D_STATE 16
#define DT_RANK 48
#define NSUBJ   8
#define LMAX    1024
#define NROWS   (NSUBJ * LMAX)       // 8192 padded rows
#define TOTAL_TOK 6848
#define DBL_W   (DT_RANK + 2 * D_STATE)   // 80
#define EPS_F   1e-5f

// GEMM tiling
#define KC    32                     // K chunk staged in LDS
#define LDS_STRIDE (KC + 4)          // 36 floats: 144B rows -> 16B-aligned, bank-spread

__constant__ int c_LEN[NSUBJ]   = {1024, 896, 768, 1024, 512, 640, 960, 1024};
__constant__ int c_OFF[NSUBJ+1] = {0, 1024, 1920, 2688, 3712, 4224, 4864, 5824, 6848};

typedef __attribute__((ext_vector_type(2))) float v2f;
typedef __attribute__((ext_vector_type(8))) float v8f;

// ---------------- CDNA5 async copy helpers ----------------
__device__ __forceinline__ void async_b128_to_lds(uint32_t lds_byte_off, const float* gaddr) {
    // GLOBAL_LOAD_ASYNC_TO_LDS_B128: per-lane 16B global -> LDS, tracked by ASYNCcnt.
    asm volatile("global_load_async_to_lds_b128 %0, %1, off"
                 :: "v"(lds_byte_off), "v"(gaddr) : "memory");
}
__device__ __forceinline__ void wait_asynccnt0() {
    asm volatile("s_wait_asynccnt 0x0" ::: "memory");
}

// ---------------- helpers ----------------
__device__ __forceinline__ float block_reduce_sum256(float v, float* red) {
    const int tid = threadIdx.x;
    red[tid] = v;
    __syncthreads();
    for (int off = 128; off > 0; off >>= 1) {
        if (tid < off) red[tid] += red[tid + off];
        __syncthreads();
    }
    float r = red[0];
    __syncthreads();
    return r;
}

__device__ __forceinline__ float silu_f(float z) {
    return z / (1.0f + __expf(-z));
}

// ---------------- kernels ----------------

// Embedding gather + RMSNorm(in_norm_w) + pack to padded [8,1024,768]; pads -> 0.
__global__ __launch_bounds__(256)
void k_embed_norm_pack(const int* __restrict__ tokens, const float* __restrict__ embed,
                       const float* __restrict__ w, float* __restrict__ out) {
    __shared__ float red[256];
    const int r = blockIdx.x, s = r >> 10, pos = r & (LMAX - 1);
    const int tid = threadIdx.x;
    float v0 = 0.f, v1 = 0.f, v2 = 0.f;
    if (pos < c_LEN[s]) {
        const int tok = tokens[c_OFF[s] + pos];
        const float* src = embed + (size_t)tok * D_MODEL;
        v0 = src[tid]; v1 = src[tid + 256]; v2 = src[tid + 512];
    }
    const float ss = block_reduce_sum256(v0 * v0 + v1 * v1 + v2 * v2, red);
    const float sc = rsqrtf(ss * (1.0f / D_MODEL) + EPS_F);
    float* dst = out + (size_t)r * D_MODEL;
    dst[tid]       = v0 * sc * w[tid];
    dst[tid + 256] = v1 * sc * w[tid + 256];
    dst[tid + 512] = v2 * sc * w[tid + 512];
}

// RMSNorm over 768 columns, one block per row.
__global__ __launch_bounds__(256)
void k_rmsnorm(const float* __restrict__ in, const float* __restrict__ w,
               float* __restrict__ out) {
    __shared__ float red[256];
    const int tid = threadIdx.x;
    const float* src = in + (size_t)blockIdx.x * D_MODEL;
    const float v0 = src[tid], v1 = src[tid + 256], v2 = src[tid + 512];
    const float ss = block_reduce_sum256(v0 * v0 + v1 * v1 + v2 * v2, red);
    const float sc = rsqrtf(ss * (1.0f / D_MODEL) + EPS_F);
    float* dst = out + (size_t)blockIdx.x * D_MODEL;
    dst[tid]       = v0 * sc * w[tid];
    dst[tid + 256] = v1 * sc * w[tid + 256];
    dst[tid + 512] = v2 * sc * w[tid + 512];
}

// C[M,N] (=/+=) A[M,K] * W[N,K]^T via V_WMMA_F32_16X16X4_F32.
// Block tile: 128 rows x (NT*16) cols, 8 waves (one 16-row strip each).
// A/W K-chunks (KC=32) staged into double-buffered LDS with async B128 copies;
// chunk k+1 DMA overlaps chunk k compute. Caller guarantees exact coverage:
// M = gridDim.y*128, all NT tiles at n_off + blockIdx.x*NT*16 inside N, K % KC == 0.
template<int NT, int ACC>
__global__ __launch_bounds__(256)
void k_gemm(const float* __restrict__ A, const float* __restrict__ W,
            float* __restrict__ C, int K, int lda, int ldw, int ldc, int n_off) {
    __shared__ float sA[2][128 * LDS_STRIDE];
    __shared__ float sW[2][NT * 16 * LDS_STRIDE];

    const int tid  = threadIdx.x;
    const int lane = tid & 31;
    const int wave = tid >> 5;
    const int row0 = blockIdx.y * 128;
    const int l16  = lane & 15;
    const int kl   = (lane >> 4) << 1;   // 0 or 2
    const int nbase = n_off + blockIdx.x * (NT * 16);

    // D/C tile mapping: lane covers col nbase+j*16+l16, rows cm..cm+7
    const int cm = row0 + wave * 16 + ((lane >> 4) << 3);
    v8f acc[NT];
#pragma unroll
    for (int j = 0; j < NT; ++j) {
        const int n = nbase + j * 16 + l16;
#pragma unroll
        for (int v = 0; v < 8; ++v)
            acc[j][v] = ACC ? C[(size_t)(cm + v) * ldc + n] : 0.0f;
    }

    auto prefetch = [&](int k0, int buf) {
        // A tile: 128 rows x KC, as 16B chunks (4 floats): 8 chunks/row, 1024 total
        constexpr int ACH = 128 * (KC / 4);
#pragma unroll
        for (int c = tid; c < ACH; c += 256) {
            const int r  = c >> 3;
            const int kc = (c & 7) << 2;
            const uint32_t lo =
                (uint32_t)(uintptr_t)(&sA[buf][r * LDS_STRIDE + kc]);
            async_b128_to_lds(lo, A + (size_t)(row0 + r) * lda + (k0 + kc));
        }
        // W tile: NT*16 rows x KC
        constexpr int WCH = NT * 16 * (KC / 4);
#pragma unroll
        for (int c = tid; c < WCH; c += 256) {
            const int n  = c >> 3;
            const int kc = (c & 7) << 2;
            const uint32_t lo =
                (uint32_t)(uintptr_t)(&sW[buf][n * LDS_STRIDE + kc]);
            async_b128_to_lds(lo, W + (size_t)(nbase + n) * ldw + (k0 + kc));
        }
    };

    prefetch(0, 0);
    const int nch = K / KC;
    for (int kc = 0; kc < nch; ++kc) {
        wait_asynccnt0();      // this wave's chunk-kc copies done
        __syncthreads();       // all waves' copies done; all done reading buf[(kc+1)&1]
        if (kc + 1 < nch) prefetch((kc + 1) * KC, (kc + 1) & 1);

        const float* a_base = &sA[kc & 1][(wave * 16 + l16) * LDS_STRIDE + kl];
        const float* w_base = &sW[kc & 1][l16 * LDS_STRIDE + kl];
#pragma unroll
        for (int k4 = 0; k4 < KC; k4 += 4) {
            const v2f a = *(const v2f*)(a_base + k4);
#pragma unroll
            for (int j = 0; j < NT; ++j) {
                const v2f b = *(const v2f*)(w_base + j * 16 * LDS_STRIDE + k4);
                acc[j] = __builtin_amdgcn_wmma_f32_16x16x4_f32(
                    false, a, false, b, (short)0, acc[j], false, false);
            }
        }
    }

#pragma unroll
    for (int j = 0; j < NT; ++j) {
        const int n = nbase + j * 16 + l16;
#pragma unroll
        for (int v = 0; v < 8; ++v)
            C[(size_t)(cm + v) * ldc + n] = acc[j][v];
    }
}

// Causal depthwise conv (K=4) over x-half of xz + bias + SiLU -> xconv.
__global__ __launch_bounds__(256)
void k_conv_silu(const float* __restrict__ xz, const float* __restrict__ cw,
                 const float* __restrict__ cb, float* __restrict__ xout) {
    const size_t idx = (size_t)blockIdx.x * 256 + threadIdx.x;   // over 8192*1536
    const int c = (int)(idx % D_INNER);
    const int r = (int)(idx / D_INNER);
    const int s = r >> 10, pos = r & (LMAX - 1);
    float acc = cb[c];
    const float* w4 = cw + c * 4;
#pragma unroll
    for (int k = 0; k < 4; ++k) {
        const int tp = pos - 3 + k;
        if (tp >= 0)
            acc += xz[((size_t)((s << 10) + tp)) * (2 * D_INNER) + c] * w4[k];
    }
    xout[idx] = silu_f(acc);
}

// dt = softplus(dbl[:, :48] @ dt_proj_w^T + b); one block per row.
__global__ __launch_bounds__(256)
void k_dtproj(const float* __restrict__ dbl, const float* __restrict__ wt,
              const float* __restrict__ bt, float* __restrict__ dt) {
    __shared__ float sd[DT_RANK];
    const int r = blockIdx.x;
    if (threadIdx.x < DT_RANK) sd[threadIdx.x] = dbl[(size_t)r * DBL_W + threadIdx.x];
    __syncthreads();
#pragma unroll
    for (int j = 0; j < D_INNER / 256; ++j) {
        const int d = threadIdx.x + j * 256;
        float acc = bt[d];
        const float* wr = wt + (size_t)d * DT_RANK;
#pragma unroll
        for (int i = 0; i < DT_RANK; ++i) acc += sd[i] * wr[i];
        dt[(size_t)r * D_INNER + d] = (acc > 20.f) ? acc : log1pf(__expf(acc));
    }
}

// Selective scan + D skip + SiLU(z) gate. Block = 256 channels of one subject.
// Pads (t >= L) written as zero so the residual GEMM stays clean.
__global__ __launch_bounds__(256)
void k_scan(const float* __restrict__ x, const float* __restrict__ dtb,
            const float* __restrict__ dbl, const float* __restrict__ xz,
            const float* __restrict__ A_log, const float* __restrict__ Dp,
            float* __restrict__ y) {
    const int blk = blockIdx.x;                 // 48 = 8 subjects * 6 chunks
    const int s = blk / (D_INNER / 256);
    const int chunk = blk % (D_INNER / 256);
    const int d = chunk * 256 + threadIdx.x;
    const int L = c_LEN[s];
    const size_t rowbase = (size_t)s << 10;

    float A[D_STATE], h[D_STATE];
#pragma unroll
    for (int n = 0; n < D_STATE; ++n) {
        A[n] = -__expf(A_log[(size_t)d * D_STATE + n]);
        h[n] = 0.0f;
    }
    const float Dv = Dp[d];
    __shared__ float sB[D_STATE], sC[D_STATE];

    for (int t = 0; t < L; ++t) {
        const size_t row = rowbase + t;
        if (threadIdx.x < D_STATE)
            sB[threadIdx.x] = dbl[row * DBL_W + DT_RANK + threadIdx.x];
        else if (threadIdx.x < 2 * D_STATE)
            sC[threadIdx.x - D_STATE] = dbl[row * DBL_W + DT_RANK + threadIdx.x];
        __syncthreads();

        const float xt  = x[row * D_INNER + d];
        const float dtt = dtb[row * D_INNER + d];
        const float dx  = dtt * xt;
        float yv = 0.0f;
#pragma unroll
        for (int n = 0; n < D_STATE; ++n) {
            const float dA = __expf(dtt * A[n]);
            h[n] = dA * h[n] + dx * sB[n];
            yv += h[n] * sC[n];
        }
        yv += xt * Dv;
        const float z = xz[row * (2 * D_INNER) + D_INNER + d];
        y[row * D_INNER + d] = yv * silu_f(z);
        __syncthreads();
    }
    for (int t = L; t < LMAX; ++t)
        y[(rowbase + t) * D_INNER + d] = 0.0f;
}

// Final: rmsnorm(rmsnorm(h, norm_f_w), out_norm_w), unpadded to token order.
__global__ __launch_bounds__(256)
void k_final(const float* __restrict__ h, const float* __restrict__ wf,
             const float* __restrict__ wo, float* __restrict__ out) {
    __shared__ float red[256];
    const int t = blockIdx.x;
    int s = 0;
#pragma unroll
    for (int i = 1; i < NSUBJ; ++i) if (t >= c_OFF[i]) s = i;
    const int pos = t - c_OFF[s];
    const float* src = h + (size_t)((s << 10) + pos) * D_MODEL;
    const int tid = threadIdx.x;
    const float v0 = src[tid], v1 = src[tid + 256], v2 = src[tid + 512];
    const float ss = block_reduce_sum256(v0 * v0 + v1 * v1 + v2 * v2, red);
    const float sc = rsqrtf(ss * (1.0f / D_MODEL) + EPS_F);
    const float a0 = v0 * sc * wf[tid];
    const float a1 = v1 * sc * wf[tid + 256];
    const float a2 = v2 * sc * wf[tid + 512];
    const float ss2 = block_reduce_sum256(a0 * a0 + a1 * a1 + a2 * a2, red);
    const float sc2 = rsqrtf(ss2 * (1.0f / D_MODEL) + EPS_F);
    float* dst = out + (size_t)t * D_MODEL;
    dst[tid]       = a0 * sc2 * wo[tid];
    dst[tid + 256] = a1 * sc2 * wo[tid + 256];
    dst[tid + 512] = a2 * sc2 * wo[tid + 512];
}

// ---------------- launcher ----------------
extern "C" void kernel_launch(void* const* d_in, const int* in_sizes, int n_in,
                              void* d_out, int out_size, void* d_ws, size_t ws_size,
                              hipStream_t stream) {
    (void)in_sizes; (void)n_in; (void)out_size; (void)ws_size;
    const int*   tokens     = (const int*)d_in[0];
    const float* embed      = (const float*)d_in[1];
    const float* in_norm_w  = (const float*)d_in[2];
    const float* out_norm_w = (const float*)d_in[3];
    const float* norm_w     = (const float*)d_in[4];
    const float* in_proj_w  = (const float*)d_in[5];
    const float* conv_w     = (const float*)d_in[6];
    const float* conv_b     = (const float*)d_in[7];
    const float* x_proj_w   = (const float*)d_in[8];
    const float* dt_proj_w  = (const float*)d_in[9];
    const float* dt_proj_b  = (const float*)d_in[10];
    const float* A_log      = (const float*)d_in[11];
    const float* D_param    = (const float*)d_in[12];
    const float* out_proj_w = (const float*)d_in[13];
    const float* norm_f_w   = (const float*)d_in[14];

    float* ws    = (float*)d_ws;
    float* h_pad = ws;  ws += (size_t)NROWS * D_MODEL;      // residual stream (padded)
    float* tnorm = ws;  ws += (size_t)NROWS * D_MODEL;
    float* xz    = ws;  ws += (size_t)NROWS * 2 * D_INNER;  // [x | z]
    float* xconv = ws;  ws += (size_t)NROWS * D_INNER;
    float* dblb  = ws;  ws += (size_t)NROWS * DBL_W;
    float* dtb   = ws;  ws += (size_t)NROWS * D_INNER;
    float* ybuf  = ws;

    k_embed_norm_pack<<<NROWS, 256, 0, stream>>>(tokens, embed, in_norm_w, h_pad);

    for (int l = 0; l < 2; ++l) {
        k_rmsnorm<<<NROWS, 256, 0, stream>>>(h_pad, norm_w + l * D_MODEL, tnorm);

        // xz = tnorm @ in_proj_w^T : [8192 x 3072], K=768
        k_gemm<4, 0><<<dim3(48, 64), 256, 0, stream>>>(
            tnorm, in_proj_w + (size_t)l * 2 * D_INNER * D_MODEL, xz,
            D_MODEL, D_MODEL, D_MODEL, 2 * D_INNER, 0);

        k_conv_silu<<<(NROWS * D_INNER) / 256, 256, 0, stream>>>(
            xz, conv_w + l * D_INNER * 4, conv_b + l * D_INNER, xconv);

        // dbl = xconv @ x_proj_w^T : [8192 x 80], K=1536 (4 tiles + 1 remainder tile)
        k_gemm<4, 0><<<dim3(1, 64), 256, 0, stream>>>(
            xconv, x_proj_w + (size_t)l * DBL_W * D_INNER, dblb,
            D_INNER, D_INNER, D_INNER, DBL_W, 0);
        k_gemm<1, 0><<<dim3(1, 64), 256, 0, stream>>>(
            xconv, x_proj_w + (size_t)l * DBL_W * D_INNER, dblb,
            D_INNER, D_INNER, D_INNER, DBL_W, 64);

        k_dtproj<<<NROWS, 256, 0, stream>>>(
            dblb, dt_proj_w + (size_t)l * D_INNER * DT_RANK, dt_proj_b + l * D_INNER, dtb);

        k_scan<<<NSUBJ * (D_INNER / 256), 256, 0, stream>>>(
            xconv, dtb, dblb, xz, A_log + (size_t)l * D_INNER * D_STATE,
            D_param + l * D_INNER, ybuf);

        // h_pad += ybuf @ out_proj_w^T : [8192 x 768], K=1536
        k_gemm<4, 1><<<dim3(12, 64), 256, 0, stream>>>(
            ybuf, out_proj_w + (size_t)l * D_MODEL * D_INNER, h_pad,
            D_INNER, D_INNER, D_INNER, D_MODEL, 0);
    }

    k_final<<<TOTAL_TOK, 256, 0, stream>>>(h_pad, norm_f_w, out_norm_w, (float*)d_out);
}